// GINMultiTask_472446402721
// MI455X (gfx1250) — compile-verified
//
#include <hip/hip_runtime.h>
#include <hip/hip_bf16.h>

typedef float v2f __attribute__((ext_vector_type(2)));
typedef float v8f __attribute__((ext_vector_type(8)));

#define NN 50000
#define NE 1600000
#define DIM 128
#define SH 64
#define NG 2048
#define PITCH 132   // 128 + 4 dwords pad -> bank stride 4, conflict-free column access

// ---------------------------------------------------------------- utilities
__global__ void zero_kernel(float* __restrict__ p, long n) {
    long i = (long)blockIdx.x * blockDim.x + threadIdx.x;
    if (i < n) p[i] = 0.0f;
}

// ------------------------------------------------------- edge scatter (agg += feat[src] at dst)
// one 32-thread group per edge; each lane moves a float4 chunk (coalesced 16B loads)
__global__ __launch_bounds__(256)
void scatter_add_kernel(const float* __restrict__ feat,
                        const int* __restrict__ src,
                        const int* __restrict__ dst,
                        float* __restrict__ agg) {
    long gid = (long)blockIdx.x * blockDim.x + threadIdx.x;
    long e = gid >> 5;
    int  c = (int)(gid & 31);
    if (e >= NE) return;
    int s = src[e];
    int d = dst[e];
    const float4 v = *(const float4*)(feat + (long)s * DIM + c * 4);
    float* o = agg + (long)d * DIM + c * 4;
    atomicAdd(o + 0, v.x);
    atomicAdd(o + 1, v.y);
    atomicAdd(o + 2, v.z);
    atomicAdd(o + 3, v.w);
}

// ------------------------------------------------------- fused GIN MLP:
// out = relu( relu( (x + agg) @ Wa + ba ) @ Wb + bb )
// block = 256 threads (8 waves); 16 rows per block; wave w owns N-tile [16w, 16w+16)
__global__ __launch_bounds__(256)
void gin_mlp_kernel(const float* __restrict__ xin, const float* __restrict__ agg,
                    const float* __restrict__ Wa,  const float* __restrict__ ba,
                    const float* __restrict__ Wb,  const float* __restrict__ bb,
                    float* __restrict__ out) {
    __shared__ float tA[16 * PITCH];   // input tile (x + agg)
    __shared__ float tT[16 * PITCH];   // intermediate relu tile

    const int tid  = threadIdx.x;
    const int wave = tid >> 5;
    const int lane = tid & 31;
    const long rowbase = (long)blockIdx.x * 16;   // 50000/16 = 3125 exactly, no tail

    // cooperative load of h = x + agg (16 x 128)
    for (int i = tid; i < 16 * DIM; i += 256) {
        int r = i >> 7, c = i & 127;
        long g = (rowbase + r) * DIM + c;
        tA[r * PITCH + c] = xin[g] + agg[g];
    }
    __syncthreads();

    const int n0    = wave * 16;
    const int nlane = n0 + (lane & 15);
    const int arow  = lane & 15;
    const int koff  = (lane >> 4) * 2;    // lanes 0-15: K+0,K+1 ; lanes 16-31: K+2,K+3

    // ---- stage 1: tT = relu(tA @ Wa + ba)
    v8f acc = {0.f,0.f,0.f,0.f,0.f,0.f,0.f,0.f};
    for (int k0 = 0; k0 < DIM; k0 += 4) {
        v2f a = *(const v2f*)&tA[arow * PITCH + k0 + koff];
        v2f b;
        b.x = Wa[(k0 + 0 + koff) * DIM + nlane];
        b.y = Wa[(k0 + 1 + koff) * DIM + nlane];
        acc = __builtin_amdgcn_wmma_f32_16x16x4_f32(false, a, false, b,
                                                    (short)0, acc, false, false);
    }
    {
        float bias = ba[nlane];
        #pragma unroll
        for (int r = 0; r < 8; r++) {
            int row = (lane < 16) ? r : (r + 8);     // C layout: VGPR r -> M=r / M=r+8
            float v = acc[r] + bias;
            tT[row * PITCH + nlane] = v > 0.f ? v : 0.f;
        }
    }
    __syncthreads();

    // ---- stage 2: out = relu(tT @ Wb + bb)
    v8f acc2 = {0.f,0.f,0.f,0.f,0.f,0.f,0.f,0.f};
    for (int k0 = 0; k0 < DIM; k0 += 4) {
        v2f a = *(const v2f*)&tT[arow * PITCH + k0 + koff];
        v2f b;
        b.x = Wb[(k0 + 0 + koff) * DIM + nlane];
        b.y = Wb[(k0 + 1 + koff) * DIM + nlane];
        acc2 = __builtin_amdgcn_wmma_f32_16x16x4_f32(false, a, false, b,
                                                     (short)0, acc2, false, false);
    }
    {
        float bias = bb[nlane];
        #pragma unroll
        for (int r = 0; r < 8; r++) {
            int row = (lane < 16) ? r : (r + 8);
            float v = acc2[r] + bias;
            out[(rowbase + row) * DIM + nlane] = v > 0.f ? v : 0.f;
        }
    }
}

// ------------------------------------------------------- graph pooling
__global__ __launch_bounds__(256)
void pool_sum_kernel(const float* __restrict__ h, const int* __restrict__ batch,
                     float* __restrict__ sums) {
    long gid = (long)blockIdx.x * blockDim.x + threadIdx.x;
    long n = gid >> 5;
    int  c = (int)(gid & 31);
    if (n >= NN) return;
    int g = batch[n];
    const float4 v = *(const float4*)(h + n * DIM + c * 4);
    float* o = sums + (long)g * DIM + c * 4;
    atomicAdd(o + 0, v.x);
    atomicAdd(o + 1, v.y);
    atomicAdd(o + 2, v.z);
    atomicAdd(o + 3, v.w);
}

__global__ __launch_bounds__(256)
void pool_count_kernel(const int* __restrict__ batch, float* __restrict__ counts) {
    long n = (long)blockIdx.x * blockDim.x + threadIdx.x;
    if (n < NN) atomicAdd(&counts[batch[n]], 1.0f);
}

// ------------------------------------------------------- pooled MLP + 3 heads
// one 64-thread block per graph
__global__ __launch_bounds__(64)
void head_kernel(const float* __restrict__ sums, const float* __restrict__ counts,
                 const float* __restrict__ Ws,  const float* __restrict__ bs,
                 const float* __restrict__ WlS, const float* __restrict__ blS,
                 const float* __restrict__ WlP, const float* __restrict__ blP,
                 const float* __restrict__ WnR, const float* __restrict__ bnR,
                 float* __restrict__ outp) {
    __shared__ float pooled[DIM];
    __shared__ float gsh[SH];
    const int g = blockIdx.x;
    const int t = threadIdx.x;

    float cnt = counts[g];
    cnt = cnt > 1.0f ? cnt : 1.0f;
    float inv = 1.0f / cnt;
    for (int c = t; c < DIM; c += 64) pooled[c] = sums[(long)g * DIM + c] * inv;
    __syncthreads();

    float a = bs[t];
    for (int k = 0; k < DIM; k++) a += pooled[k] * Ws[k * SH + t];
    gsh[t] = a > 0.f ? a : 0.f;
    __syncthreads();

    if (t < 3) {
        const float* W = (t == 0) ? WlS : (t == 1) ? WlP : WnR;
        float s = (t == 0) ? blS[0] : (t == 1) ? blP[0] : bnR[0];
        for (int k = 0; k < SH; k++) s += gsh[k] * W[k];
        outp[t * NG + g] = s;
    }
}

// ----------------------------------------------------------------------------
extern "C" void kernel_launch(void* const* d_in, const int* in_sizes, int n_in,
                              void* d_out, int out_size, void* d_ws, size_t ws_size,
                              hipStream_t stream) {
    const float* x     = (const float*)d_in[0];
    const int*   ei    = (const int*)d_in[1];     // (2, NE): [src | dst]
    const int*   src   = ei;
    const int*   dst   = ei + NE;
    const int*   batch = (const int*)d_in[2];
    const float* W1a = (const float*)d_in[3];  const float* b1a = (const float*)d_in[4];
    const float* W1b = (const float*)d_in[5];  const float* b1b = (const float*)d_in[6];
    const float* W2a = (const float*)d_in[7];  const float* b2a = (const float*)d_in[8];
    const float* W2b = (const float*)d_in[9];  const float* b2b = (const float*)d_in[10];
    const float* Ws  = (const float*)d_in[11]; const float* bs  = (const float*)d_in[12];
    const float* WlS = (const float*)d_in[13]; const float* blS = (const float*)d_in[14];
    const float* WlP = (const float*)d_in[15]; const float* blP = (const float*)d_in[16];
    const float* WnR = (const float*)d_in[17]; const float* bnR = (const float*)d_in[18];
    float* outp = (float*)d_out;

    // workspace layout
    float* agg    = (float*)d_ws;                 // NN*DIM
    float* h1     = agg  + (long)NN * DIM;        // NN*DIM
    float* h2     = h1   + (long)NN * DIM;        // NN*DIM
    float* sums   = h2   + (long)NN * DIM;        // NG*DIM
    float* counts = sums + (long)NG * DIM;        // NG

    const long nagg  = (long)NN * DIM;            // 6,400,000
    const long npool = (long)NG * DIM + NG;       // sums + counts contiguous
    const long eth   = (long)NE * 32;             // 51,200,000 scatter threads
    const long pth   = (long)NN * 32;

    // conv layer 1
    zero_kernel<<<(unsigned)((nagg + 255) / 256), 256, 0, stream>>>(agg, nagg);
    zero_kernel<<<(unsigned)((npool + 255) / 256), 256, 0, stream>>>(sums, npool);
    scatter_add_kernel<<<(unsigned)(eth / 256), 256, 0, stream>>>(x, src, dst, agg);
    gin_mlp_kernel<<<NN / 16, 256, 0, stream>>>(x, agg, W1a, b1a, W1b, b1b, h1);

    // conv layer 2
    zero_kernel<<<(unsigned)((nagg + 255) / 256), 256, 0, stream>>>(agg, nagg);
    scatter_add_kernel<<<(unsigned)(eth / 256), 256, 0, stream>>>(h1, src, dst, agg);
    gin_mlp_kernel<<<NN / 16, 256, 0, stream>>>(h1, agg, W2a, b2a, W2b, b2b, h2);

    // mean pool + heads
    pool_sum_kernel<<<(unsigned)((pth + 255) / 256), 256, 0, stream>>>(h2, batch, sums);
    pool_count_kernel<<<(unsigned)((NN + 255) / 256), 256, 0, stream>>>(batch, counts);
    head_kernel<<<NG, 64, 0, stream>>>(sums, counts, Ws, bs, WlS, blS, WlP, blP, WnR, bnR, outp);
}